// IncrementalMultiheadAttention_44573170598139
// MI455X (gfx1250) — compile-verified
//
#include <hip/hip_runtime.h>

// ---------------------------------------------------------------------------
// Incremental MHA decode step for MI455X (gfx1250, wave32).
//   H=1024, 16 heads x 64 dim, batch 32, t=1, prev_len=4095 (S=4096).
// Stage 1: q/k/v projections via V_WMMA_F32_16X16X4_F32 (exact fp32 path).
// Stage 2: per-head streaming attention (memory-bound: 1.07 GB KV @ 23.3 TB/s),
//          non-temporal 128b loads, register-resident scores, LDS softmax.
// Stage 3: output projection via the same WMMA GEMM.
// ---------------------------------------------------------------------------

#define HIDDEN    1024
#define NUM_HEADS 16
#define HEAD_DIM  64
#define BATCH     32
#define PREV_LEN  4095
#define S_TOT     4096
#define SCALING   0.125f                 // HEAD_DIM ** -0.5
#define BIG_F     3.402823466e38f        // jnp.finfo(f32).max

typedef float v2f __attribute__((ext_vector_type(2)));
typedef float v4f __attribute__((ext_vector_type(4)));
typedef float v8f __attribute__((ext_vector_type(8)));

// ---------------------------------------------------------------------------
// Out[m,n] = (sum_k X[m,k] * W[n,k] + bias[n]) * scale        (X @ W^T + b)
// M=32, N=K=1024. One wave computes one 16x16 tile; 4 waves per block.
// A-frag (16x4 f32, 2 VGPRs): lane&15 -> M row, lane>>4 selects K pair {0,1}/{2,3}.
// B-frag (4x16 f32, 2 VGPRs): lane&15 -> N col, lane>>4 selects K pair.
// C/D (8 VGPRs): VGPR r, lanes 0-15 -> M=r, lanes 16-31 -> M=r+8, N=lane&15.
// ---------------------------------------------------------------------------
__global__ __launch_bounds__(128) void gemm_xwT_wmma(
    const float* __restrict__ X, const float* __restrict__ W,
    const float* __restrict__ bias, float* __restrict__ Out, float scale)
{
  constexpr int K = HIDDEN, N = HIDDEN;
  const int wave = threadIdx.x >> 5;
  const int lane = threadIdx.x & 31;
  const int tile = blockIdx.x * 4 + wave;           // 128 tiles total (2 x 64)
  const int tilesN = N >> 4;
  const int m0 = (tile / tilesN) << 4;
  const int n0 = (tile % tilesN) << 4;
  const int lh = lane >> 4;                         // 0: K={0,1}, 1: K={2,3}
  const int ll = lane & 15;

  const float* ap = X + (size_t)(m0 + ll) * K + (lh << 1);
  const float* bp = W + (size_t)(n0 + ll) * K + (lh << 1);

  v8f c = {0.f, 0.f, 0.f, 0.f, 0.f, 0.f, 0.f, 0.f};
#pragma unroll 8
  for (int k0 = 0; k0 < K; k0 += 4) {
    v2f a = *reinterpret_cast<const v2f*>(ap + k0);
    v2f b = *reinterpret_cast<const v2f*>(bp + k0);
    // (neg_a, A, neg_b, B, c_mod, C, reuse_a, reuse_b)
    c = __builtin_amdgcn_wmma_f32_16x16x4_f32(false, a, false, b,
                                              (short)0, c, false, false);
  }

  const float bb = bias[n0 + ll];
#pragma unroll
  for (int r = 0; r < 8; ++r) {
    const int row = r + (lh << 3);
    Out[(size_t)(m0 + row) * N + n0 + ll] = (c[r] + bb) * scale;
  }
}

// ---------------------------------------------------------------------------
// Decode attention: one workgroup (256 thr) per head (512 heads).
// Thread owns 16 sequence positions (s = tid + i*256); scores stay in VGPRs.
// Softmax via two 256-wide LDS tree reductions; V accumulated per-thread then
// combined with LDS float atomics.
// ---------------------------------------------------------------------------
__global__ __launch_bounds__(256) void attn_decode(
    const float* __restrict__ qws, const float* __restrict__ kws,
    const float* __restrict__ vws,
    const float* __restrict__ prev_key, const float* __restrict__ prev_value,
    const int* __restrict__ order, const int* __restrict__ mask,
    float* __restrict__ attn_out)
{
  __shared__ float red[256];
  __shared__ float out_sh[HEAD_DIM];

  const int tid  = threadIdx.x;
  const int head = blockIdx.x;
  const int b    = head >> 4;
  const int h    = head & 15;
  const int src  = order[b];   // beam reorder of the cache

  const float* kprev = prev_key   + (((size_t)src * NUM_HEADS + h) * PREV_LEN) * HEAD_DIM;
  const float* vprev = prev_value + (((size_t)src * NUM_HEADS + h) * PREV_LEN) * HEAD_DIM;
  const float* qn = qws + (size_t)b * HIDDEN + h * HEAD_DIM;   // already *SCALING
  const float* kn = kws + (size_t)b * HIDDEN + h * HEAD_DIM;   // new step (not reordered)
  const float* vn = vws + (size_t)b * HIDDEN + h * HEAD_DIM;

  v4f qv[16];
#pragma unroll
  for (int j = 0; j < 16; ++j)
    qv[j] = *reinterpret_cast<const v4f*>(qn + 4 * j);

  // ---- scores: q . k[s], additive -INF mask ----
  float scr[16];
  float lmax = -BIG_F;
#pragma unroll
  for (int i = 0; i < 16; ++i) {
    const int s = tid + (i << 8);
    const float* kp = (s < PREV_LEN) ? (kprev + (size_t)s * HEAD_DIM) : kn;
    float dot = 0.f;
#pragma unroll
    for (int j = 0; j < 16; ++j) {
      const v4f kk = __builtin_nontemporal_load(reinterpret_cast<const v4f*>(kp) + j);
      const v4f p = qv[j] * kk;
      dot += p.x + p.y + p.z + p.w;
    }
    dot -= BIG_F * (float)mask[s];
    scr[i] = dot;
    lmax = fmaxf(lmax, dot);
  }

  // ---- max reduce ----
  red[tid] = lmax;
  __syncthreads();
#pragma unroll
  for (int st = 128; st > 0; st >>= 1) {
    if (tid < st) red[tid] = fmaxf(red[tid], red[tid + st]);
    __syncthreads();
  }
  const float mx = red[0];
  __syncthreads();

  // ---- exp + sum reduce ----
  float lsum = 0.f;
#pragma unroll
  for (int i = 0; i < 16; ++i) {
    scr[i] = __expf(scr[i] - mx);
    lsum += scr[i];
  }
  red[tid] = lsum;
  __syncthreads();
#pragma unroll
  for (int st = 128; st > 0; st >>= 1) {
    if (tid < st) red[tid] += red[tid + st];
    __syncthreads();
  }
  const float inv = 1.0f / red[0];

  // ---- attn = sum_s w[s] * v[s,:] ----
  if (tid < HEAD_DIM) out_sh[tid] = 0.f;
  __syncthreads();

  v4f acc[16];
#pragma unroll
  for (int j = 0; j < 16; ++j) acc[j] = (v4f){0.f, 0.f, 0.f, 0.f};

#pragma unroll
  for (int i = 0; i < 16; ++i) {
    const int s = tid + (i << 8);
    const float w = scr[i] * inv;
    const v4f wv = {w, w, w, w};
    const float* vp = (s < PREV_LEN) ? (vprev + (size_t)s * HEAD_DIM) : vn;
#pragma unroll
    for (int j = 0; j < 16; ++j) {
      const v4f vvv = __builtin_nontemporal_load(reinterpret_cast<const v4f*>(vp) + j);
      acc[j] += wv * vvv;
    }
  }

#pragma unroll
  for (int j = 0; j < 16; ++j) {
    atomicAdd(&out_sh[4 * j + 0], acc[j].x);
    atomicAdd(&out_sh[4 * j + 1], acc[j].y);
    atomicAdd(&out_sh[4 * j + 2], acc[j].z);
    atomicAdd(&out_sh[4 * j + 3], acc[j].w);
  }
  __syncthreads();

  if (tid < HEAD_DIM)
    attn_out[(size_t)b * HIDDEN + h * HEAD_DIM + tid] = out_sh[tid];
}

// ---------------------------------------------------------------------------
extern "C" void kernel_launch(void* const* d_in, const int* in_sizes, int n_in,
                              void* d_out, int out_size, void* d_ws, size_t ws_size,
                              hipStream_t stream) {
  const float* q_in  = (const float*)d_in[0];
  const float* k_in  = (const float*)d_in[1];
  const float* v_in  = (const float*)d_in[2];
  const int*   mask  = (const int*)  d_in[3];
  const int*   order = (const int*)  d_in[4];
  const float* pk    = (const float*)d_in[5];
  const float* pv    = (const float*)d_in[6];
  const float* Wq = (const float*)d_in[7];
  const float* bq = (const float*)d_in[8];
  const float* Wk = (const float*)d_in[9];
  const float* bk = (const float*)d_in[10];
  const float* Wv = (const float*)d_in[11];
  const float* bv = (const float*)d_in[12];
  const float* Wo = (const float*)d_in[13];
  const float* bo = (const float*)d_in[14];
  float* out = (float*)d_out;

  float* ws  = (float*)d_ws;           // 4 x 32768 floats = 512 KB scratch
  float* qws = ws;
  float* kws = ws + 32768;
  float* vws = ws + 65536;
  float* aws = ws + 98304;

  // Stage 1: projections (q scaled by HEAD_DIM^-0.5, applied to bias too per ref)
  gemm_xwT_wmma<<<32, 128, 0, stream>>>(q_in, Wq, bq, qws, SCALING);
  gemm_xwT_wmma<<<32, 128, 0, stream>>>(k_in, Wk, bk, kws, 1.0f);
  gemm_xwT_wmma<<<32, 128, 0, stream>>>(v_in, Wv, bv, vws, 1.0f);

  // Stage 2: streaming decode attention, one WG per (batch, head)
  attn_decode<<<BATCH * NUM_HEADS, 256, 0, stream>>>(qws, kws, vws, pk, pv,
                                                     order, mask, aws);

  // Stage 3: output projection straight into d_out
  gemm_xwT_wmma<<<32, 128, 0, stream>>>(aws, Wo, bo, out, 1.0f);
}